// TensorProduct_63720134804195
// MI455X (gfx1250) — compile-verified
//
#include <hip/hip_runtime.h>

typedef float v2f __attribute__((ext_vector_type(2)));
typedef float v8f __attribute__((ext_vector_type(8)));

constexpr int N1   = 72;      // f1 channels
constexpr int N2   = 72;      // f2 channels
constexpr int MTOT = N1 * N2; // contraction length 5184
constexpr int KTOT = 5184;    // output channels
constexpr int ZTOT = 2048;    // batch
constexpr int LDP  = 76;      // padded LDS row pitch (bank-conflict-free, 16B rows)

// Block = 64 threads = 2 waves. Each wave computes a 16(z) x 32(k) output tile
// as two 16x16 f32 WMMA accumulators. Grid = (5184/64, 2048/16) = (81, 128).
__global__ __launch_bounds__(64)
void tp_wmma(const float* __restrict__ mix,   // [KTOT, N1, N2]
             const float* __restrict__ f1,    // [ZTOT, N1]
             const float* __restrict__ f2,    // [ZTOT, N2]
             float* __restrict__ out)         // [ZTOT, KTOT]
{
    __shared__ __align__(16) float s_f1[16 * LDP];
    __shared__ __align__(16) float s_f2[16 * LDP];

    const int tid  = threadIdx.x;
    const int lane = tid & 31;
    const int wave = tid >> 5;
    const int row  = lane & 15;   // A: M index / B,D: N index
    const int half = lane >> 4;   // selects K-pair {0,1} vs {2,3}

    const int z0    = blockIdx.y * 16;
    const int kwave = blockIdx.x * 64 + wave * 32;

    // Stage the 16-row f1/f2 tiles into LDS.
    for (int idx = tid; idx < 16 * N1; idx += 64) {
        int r = idx / N1;
        int c = idx - r * N1;
        s_f1[r * LDP + c] = f1[(z0 + r) * N1 + c];
        s_f2[r * LDP + c] = f2[(z0 + r) * N2 + c];
    }
    __syncthreads();

    v8f acc0 = {};
    v8f acc1 = {};

    const int koff = half * 2;  // K offset within the 4-wide WMMA step
    // B operand: lane supplies column N=row of mix rows m..m+1 (b64 per lane).
    const float* __restrict__ b0p = mix + (size_t)(kwave + row) * (size_t)MTOT + koff;
    const float* __restrict__ b1p = b0p + (size_t)16 * (size_t)MTOT;
    const float* a2p = &s_f2[row * LDP + koff];
    const float* a1p = &s_f1[row * LDP];

    for (int i = 0; i < N1; ++i) {
        const float f1v = a1p[i];   // f1[z0+row, i]; same value for lanes l and l+16
        #pragma unroll
        for (int j = 0; j < N2; j += 4) {
            // A = f1[z,i] * f2[z, j+koff .. j+koff+1]  (rank-1 P built on the fly)
            v2f f2v = *(const v2f*)(a2p + j);
            v2f a;
            a.x = f1v * f2v.x;
            a.y = f1v * f2v.y;
            v2f b0 = *(const v2f*)(b0p + j);
            v2f b1 = *(const v2f*)(b1p + j);
            acc0 = __builtin_amdgcn_wmma_f32_16x16x4_f32(
                false, a, false, b0, (short)0, acc0, false, false);
            acc1 = __builtin_amdgcn_wmma_f32_16x16x4_f32(
                false, a, false, b1, (short)0, acc1, false, false);
        }
        b0p += N2;   // advance to next i-row of mix (m += 72)
        b1p += N2;
    }

    // D layout: lanes 0-15 -> N=lane, M=r ; lanes 16-31 -> N=lane-16, M=8+r.
    float* op = out + (size_t)(z0 + half * 8) * (size_t)KTOT + (size_t)(kwave + row);
    #pragma unroll
    for (int r = 0; r < 8; ++r) {
        op[0]  = acc0[r];
        op[16] = acc1[r];
        op += KTOT;
    }
}

extern "C" void kernel_launch(void* const* d_in, const int* in_sizes, int n_in,
                              void* d_out, int out_size, void* d_ws, size_t ws_size,
                              hipStream_t stream)
{
    const float* mix = (const float*)d_in[0];  // [5184, 72, 72]
    const float* f1  = (const float*)d_in[1];  // [2048, 72]
    const float* f2  = (const float*)d_in[2];  // [2048, 72]
    float* out = (float*)d_out;                // [2048, 5184]

    dim3 grid(KTOT / 64, ZTOT / 16);  // (81, 128)
    dim3 block(64);
    tp_wmma<<<grid, block, 0, stream>>>(mix, f1, f2, out);
}